// Raycaster_82686710383144
// MI455X (gfx1250) — compile-verified
//
#include <hip/hip_runtime.h>
#include <stdint.h>

#define IMG_W 256
#define IMG_H 256
#define VOL_N 256
#define TF_R  128

__device__ __forceinline__ int imin_i(int a, int b) { return a < b ? a : b; }

struct SampleCoords {
    int   base00, base10, base01, base11;  // (y,z) plane bases
    int   x0, x1;
    float fx, fy, fz;
};

// Map world pos at parameter t to voxel coords, exactly as reference _trilinear.
__device__ __forceinline__ SampleCoords coords_at(
    float t, float rox, float roy, float roz, float rdx, float rdy, float rdz)
{
    const float SCALE = (float)VOL_N - 1.0f - 1e-4f;   // 254.9999
    const float qx = fminf(fmaxf(0.5f * (rox + t * rdx) + 0.5f, 0.0f), 1.0f) * SCALE;
    const float qy = fminf(fmaxf(0.5f * (roy + t * rdy) + 0.5f, 0.0f), 1.0f) * SCALE;
    const float qz = fminf(fmaxf(0.5f * (roz + t * rdz) + 0.5f, 0.0f), 1.0f) * SCALE;
    const float lx = floorf(qx), ly = floorf(qy), lz = floorf(qz);
    const int x0 = (int)lx, y0 = (int)ly, z0 = (int)lz;
    const int x1 = imin_i(x0 + 1, VOL_N - 1);
    const int y1 = imin_i(y0 + 1, VOL_N - 1);
    const int z1 = imin_i(z0 + 1, VOL_N - 1);
    SampleCoords c;
    c.base00 = y0 * 65536 + z0 * 256;
    c.base10 = y1 * 65536 + z0 * 256;
    c.base01 = y0 * 65536 + z1 * 256;
    c.base11 = y1 * 65536 + z1 * 256;
    c.x0 = x0; c.x1 = x1;
    c.fx = qx - lx; c.fy = qy - ly; c.fz = qz - lz;
    return c;
}

__global__ __launch_bounds__(256) void raycast256_kernel(
    const float* __restrict__ vol,   // (1,256,256,256) f32; vol[x,y,z]=v[y*65536+z*256+x]
    const float* __restrict__ tf,    // (4,128) f32
    const float* __restrict__ cam,   // look_from (3,)
    const float* __restrict__ srp,   // sampling_rate (1,)
    float* __restrict__ out)         // (4,256,256) f32
{
    __shared__ float tf_lds[TF_R * 4];   // tf_tex layout: [entry][rgba], 16B aligned

    const int tid = threadIdx.x;

    // ---- Stage TF (4,128)->(128,4) into LDS via async global->LDS DMA.
    //      512 dwords, exactly 2 per lane: straight-line, no exec-mask loop. ----
    {
        const int e0 = tid;
        const int e1 = tid + 256;
        const float* g0 = tf + (e0 & 3) * TF_R + (e0 >> 2);
        const float* g1 = tf + (e1 & 3) * TF_R + (e1 >> 2);
        const uint32_t l0 = (uint32_t)(uintptr_t)(&tf_lds[e0]);  // LDS byte offset
        const uint32_t l1 = (uint32_t)(uintptr_t)(&tf_lds[e1]);
        asm volatile("global_load_async_to_lds_b32 %0, %1, off"
                     :: "v"(l0), "v"((uint64_t)(uintptr_t)g0) : "memory");
        asm volatile("global_load_async_to_lds_b32 %0, %1, off"
                     :: "v"(l1), "v"((uint64_t)(uintptr_t)g1) : "memory");
        asm volatile("s_wait_asynccnt 0x0" ::: "memory");
    }
    __syncthreads();

    // ---- Camera setup (wave-uniform -> compiler scalarizes to s_*_f32) ----
    const float rox = cam[0], roy = cam[1], roz = cam[2];
    const float sr  = srp[0];
    const float inv_sr = 1.0f / sr;

    float vdx = -rox, vdy = -roy, vdz = -roz;
    {
        float il = 1.0f / sqrtf(vdx * vdx + vdy * vdy + vdz * vdz);
        vdx *= il; vdy *= il; vdz *= il;
    }
    // right = normalize(cross(view_dir, (0,1,0))) = normalize((-vdz, 0, vdx))
    float rgx = -vdz, rgy = 0.0f, rgz = vdx;
    {
        float il = 1.0f / sqrtf(rgx * rgx + rgy * rgy + rgz * rgz);
        rgx *= il; rgy *= il; rgz *= il;
    }
    // up = normalize(cross(right, view_dir))
    float upx = rgy * vdz - rgz * vdy;
    float upy = rgz * vdx - rgx * vdz;
    float upz = rgx * vdy - rgy * vdx;
    {
        float il = 1.0f / sqrtf(upx * upx + upy * upy + upz * upz);
        upx *= il; upy *= il; upz *= il;
    }

    // ---- Per-pixel ray ----
    const int pix = blockIdx.x * 256 + tid;
    const int w = pix & (IMG_W - 1);   // fastest within wave32 -> coalesced stores
    const int h = pix >> 8;

    const float u = ((float)w + 0.5f) / (float)IMG_W - 0.5f;
    const float v = ((float)h + 0.5f) / (float)IMG_H - 0.5f;
    const float NEAR_H = 2.0f * 0.57735026918962576f * 0.1f;  // 2*tan(30deg)*NEAR
    const float NEAR_W = NEAR_H * ((float)IMG_W / (float)IMG_H);

    float rdx = 0.1f * vdx + (u * NEAR_W) * rgx + (v * NEAR_H) * upx;
    float rdy = 0.1f * vdy + (u * NEAR_W) * rgy + (v * NEAR_H) * upy;
    float rdz = 0.1f * vdz + (u * NEAR_W) * rgz + (v * NEAR_H) * upz;
    {
        float il = 1.0f / sqrtf(rdx * rdx + rdy * rdy + rdz * rdz);
        rdx *= il; rdy *= il; rdz *= il;
    }

    // ---- Slab test against [-1,1]^3 ----
    const float dfx = 1.0f / rdx, dfy = 1.0f / rdy, dfz = 1.0f / rdz;
    const float t1 = (-1.0f - rox) * dfx, t2 = (1.0f - rox) * dfx;
    const float t3 = (-1.0f - roy) * dfy, t4 = (1.0f - roy) * dfy;
    const float t5 = (-1.0f - roz) * dfz, t6 = (1.0f - roz) * dfz;
    const float tmin = fmaxf(fmaxf(fminf(t1, t2), fminf(t3, t4)), fminf(t5, t6));
    const float tmax = fminf(fminf(fmaxf(t1, t2), fmaxf(t3, t4)), fmaxf(t5, t6));
    const bool hit = (tmax >= 0.0f) && (tmin <= tmax);
    const float tstart = fmaxf(tmin, 0.0f);
    const float ray_len = fmaxf(tmax - tstart, 0.0f);
    const float ns = fminf(fmaxf(ceilf(sr * 0.5f * ray_len * (float)VOL_N), 1.0f), 256.0f);
    const float dt = ray_len / ns;
    const int n = hit ? (int)ns : 0;

    const float4* __restrict__ tf4 = (const float4*)tf_lds;

    float accr = 0.0f, accg = 0.0f, accb = 0.0f, acca = 0.0f;

    // Software-pipelined march: coords for sample k are ready entering iteration k;
    // we compute sample k+1's coords early and prefetch its corner cacheline so
    // address math and the global gather overlap.
    SampleCoords c = coords_at(tstart + 0.5f * dt, rox, roy, roz, rdx, rdy, rdz);

    for (int k = 0; k < n; ++k) {
        // 8-voxel gather for sample k (x pairs share a cacheline)
        const float v000 = vol[c.base00 + c.x0], v100 = vol[c.base00 + c.x1];
        const float v010 = vol[c.base10 + c.x0], v110 = vol[c.base10 + c.x1];
        const float v001 = vol[c.base01 + c.x0], v101 = vol[c.base01 + c.x1];
        const float v011 = vol[c.base11 + c.x0], v111 = vol[c.base11 + c.x1];

        // Next sample's coords (reused next iteration) + prefetch its corner line.
        const float tnext = tstart + ((float)(k + 1) + 0.5f) * dt;
        const SampleCoords cn = coords_at(tnext, rox, roy, roz, rdx, rdy, rdz);
        __builtin_prefetch(vol + (cn.base00 + cn.x0), 0, 3);

        // Trilinear interpolation (exact reference formula)
        const float fx = c.fx, fy = c.fy, fz = c.fz;
        const float c00 = v000 * (1.0f - fx) + v100 * fx;
        const float c10 = v010 * (1.0f - fx) + v110 * fx;
        const float c01 = v001 * (1.0f - fx) + v101 * fx;
        const float c11 = v011 * (1.0f - fx) + v111 * fx;
        const float c0 = c00 * (1.0f - fy) + c10 * fy;
        const float c1 = c01 * (1.0f - fy) + c11 * fy;
        const float intensity = c0 * (1.0f - fz) + c1 * fz;

        // Transfer function: two 16B LDS reads (ds_load_b128)
        const float tt = fmaxf(intensity * (float)(TF_R - 1), 0.0f);
        const float lo = floorf(tt);
        const float fr = tt - lo;
        const int li = imin_i((int)lo, TF_R - 1);
        const int hi = imin_i(li + 1, TF_R - 1);
        const float4 A = tf4[li];
        const float4 B = tf4[hi];
        const float rr = A.x * (1.0f - fr) + B.x * fr;
        const float gg = A.y * (1.0f - fr) + B.y * fr;
        const float bb = A.z * (1.0f - fr) + B.z * fr;
        const float av = A.w * (1.0f - fr) + B.w * fr;

        // Opacity correction + front-to-back compositing
        const float opacity = 1.0f - powf(fmaxf(1.0f - av, 0.0f), inv_sr);
        const float wgt = (1.0f - acca) * opacity;
        accr += wgt * rr;
        accg += wgt * gg;
        accb += wgt * bb;
        acca += wgt;

        c = cn;
    }

    // out is (4, H, W): out[c*H*W + h*W + w]
    const int o = h * IMG_W + w;
    out[o]                     = accr;
    out[IMG_W * IMG_H + o]     = accg;
    out[2 * IMG_W * IMG_H + o] = accb;
    out[3 * IMG_W * IMG_H + o] = acca;
}

extern "C" void kernel_launch(void* const* d_in, const int* in_sizes, int n_in,
                              void* d_out, int out_size, void* d_ws, size_t ws_size,
                              hipStream_t stream) {
    (void)in_sizes; (void)n_in; (void)d_ws; (void)ws_size; (void)out_size;
    const float* vol = (const float*)d_in[0];
    const float* tf  = (const float*)d_in[1];
    const float* cam = (const float*)d_in[2];
    const float* sr  = (const float*)d_in[3];
    float* out = (float*)d_out;

    raycast256_kernel<<<(IMG_W * IMG_H) / 256, 256, 0, stream>>>(vol, tf, cam, sr, out);
}